// TwoConvFourClassiLayerGCN_50448685859136
// MI455X (gfx1250) — compile-verified
//
#include <hip/hip_runtime.h>

typedef __attribute__((ext_vector_type(2))) float v2f;
typedef __attribute__((ext_vector_type(8))) float v8f;

#define N_NODES  100000
#define N_EDGES  1600000
#define N_GRAPHS 512

// ---------------------------------------------------------------------------
// Degree counting: one thread per edge, float atomic increments.
// ---------------------------------------------------------------------------
__global__ void deg_kernel(const int* __restrict__ src, const int* __restrict__ dst,
                           float* __restrict__ degOut, float* __restrict__ degIn,
                           int nEdges) {
  int e = blockIdx.x * blockDim.x + threadIdx.x;
  if (e >= nEdges) return;
  atomicAdd(&degOut[src[e]], 1.0f);
  atomicAdd(&degIn [dst[e]], 1.0f);
}

// In-place deg -> clip(deg,1)^-1/2
__global__ void rsqrt_kernel(float* __restrict__ d, int n) {
  int i = blockIdx.x * blockDim.x + threadIdx.x;
  if (i >= n) return;
  d[i] = rsqrtf(fmaxf(d[i], 1.0f));
}

// ---------------------------------------------------------------------------
// SpMM layer 1: agg[dst] += x[src] * rsOut[src]   (128 features, wave/edge)
// ---------------------------------------------------------------------------
__global__ void spmm_scatter128(const float* __restrict__ x,
                                const int* __restrict__ src,
                                const int* __restrict__ dst,
                                const float* __restrict__ rsOut,
                                float* __restrict__ agg, int nEdges) {
  int wid  = (blockIdx.x * blockDim.x + threadIdx.x) >> 5;
  int lane = threadIdx.x & 31;
  if (wid >= nEdges) return;
  int s = src[wid], d = dst[wid];
  float sc = rsOut[s];
  const float* xs = x   + (size_t)s * 128;
  float*       ad = agg + (size_t)d * 128;
#pragma unroll
  for (int f = 0; f < 128; f += 32)
    atomicAdd(&ad[f + lane], xs[f + lane] * sc);
}

// SpMM layer 2: agg[dst] += t[src]   (64 features, wave/edge, no scale)
__global__ void spmm_scatter64(const float* __restrict__ t,
                               const int* __restrict__ src,
                               const int* __restrict__ dst,
                               float* __restrict__ agg, int nEdges) {
  int wid  = (blockIdx.x * blockDim.x + threadIdx.x) >> 5;
  int lane = threadIdx.x & 31;
  if (wid >= nEdges) return;
  int s = src[wid], d = dst[wid];
  const float* ts = t   + (size_t)s * 64;
  float*       ad = agg + (size_t)d * 64;
#pragma unroll
  for (int f = 0; f < 64; f += 32)
    atomicAdd(&ad[f + lane], ts[f + lane]);
}

// ---------------------------------------------------------------------------
// WMMA f32 GEMM:  C[M,N] = post( A[M,K] @ W[K,N] )
// One wave per 16x64 output strip: 4 independent V_WMMA_F32_16X16X4_F32
// accumulator chains (latency hiding via ILP) + 4x A-operand reuse.
// VGPR layouts per CDNA5 ISA 7.12.2:
//   A 16x4:  lane(l&15)=M; v0 holds K = (l>>4)*2, v1 holds K+1
//   B 4x16:  lane(l&15)=N; v0 holds row K = (l>>4)*2, v1 holds row K+1
//   C 16x16: vgpr r -> M = r + (l>>4)*8, N = l&15
// post: optional per-row scale (commutes with GEMM), optional bias, relu.
// Requires N % 64 == 0, M % 16 == 0.
// ---------------------------------------------------------------------------
__global__ void gemm_wmma_f32_x4(const float* __restrict__ A, const float* __restrict__ W,
                                 float* __restrict__ C,
                                 const float* __restrict__ rowScale,
                                 const float* __restrict__ bias,
                                 int relu, int M, int K, int N) {
  int lane   = threadIdx.x & 31;
  int wid    = (blockIdx.x * blockDim.x + threadIdx.x) >> 5;   // wave-uniform
  int strips = N >> 6;                                          // 64-wide strips
  int nWork  = (M >> 4) * strips;
  if (wid >= nWork) return;                                     // uniform -> EXEC all-1

  int m0 = (wid / strips) << 4;
  int n0 = (wid % strips) << 6;
  int lm = lane & 15;        // M-index for A, N-index for B/C
  int lh = lane >> 4;        // lane half

  const float* arow = A + (size_t)(m0 + lm) * K;
  v8f acc0 = {}, acc1 = {}, acc2 = {}, acc3 = {};

#pragma unroll 2
  for (int k0 = 0; k0 < K; k0 += 4) {
    int ka = k0 + lh * 2;
    v2f a;
    a.x = arow[ka];
    a.y = arow[ka + 1];
    const float* w0 = W + (size_t)ka       * N + n0 + lm;
    const float* w1 = W + (size_t)(ka + 1) * N + n0 + lm;
    v2f b0, b1, b2, b3;
    b0.x = w0[0];  b0.y = w1[0];
    b1.x = w0[16]; b1.y = w1[16];
    b2.x = w0[32]; b2.y = w1[32];
    b3.x = w0[48]; b3.y = w1[48];
    acc0 = __builtin_amdgcn_wmma_f32_16x16x4_f32(false, a, false, b0, (short)0, acc0, false, false);
    acc1 = __builtin_amdgcn_wmma_f32_16x16x4_f32(false, a, false, b1, (short)0, acc1, false, false);
    acc2 = __builtin_amdgcn_wmma_f32_16x16x4_f32(false, a, false, b2, (short)0, acc2, false, false);
    acc3 = __builtin_amdgcn_wmma_f32_16x16x4_f32(false, a, false, b3, (short)0, acc3, false, false);
  }

  // Epilogue: per-row scale, bias, relu; store 4 tiles.
  v8f accs[4] = {acc0, acc1, acc2, acc3};
#pragma unroll
  for (int t = 0; t < 4; ++t) {
    int col = n0 + t * 16 + lm;
    float bb = bias ? bias[col] : 0.0f;
#pragma unroll
    for (int r = 0; r < 8; ++r) {
      int row = m0 + r + lh * 8;
      float v = accs[t][r];
      if (rowScale) v *= rowScale[row];
      v += bb;
      if (relu) v = fmaxf(v, 0.0f);
      C[(size_t)row * N + col] = v;
    }
  }
}

// ---------------------------------------------------------------------------
// Pooling: sums[g] += relu(agg2[n]*rsIn[n] + b2), cnt[g] += 1   (wave/node)
// Fuses the layer-2 epilogue so h2 is never materialized.
// ---------------------------------------------------------------------------
__global__ void pool_kernel(const float* __restrict__ agg2,
                            const float* __restrict__ rsIn,
                            const float* __restrict__ b2,
                            const int* __restrict__ gid,
                            float* __restrict__ sums, float* __restrict__ cnt,
                            int nNodes) {
  int wid  = (blockIdx.x * blockDim.x + threadIdx.x) >> 5;
  int lane = threadIdx.x & 31;
  if (wid >= nNodes) return;
  int g = gid[wid];
  float sc = rsIn[wid];
  const float* an = agg2 + (size_t)wid * 64;
#pragma unroll
  for (int f = 0; f < 64; f += 32) {
    float v = fmaxf(an[f + lane] * sc + b2[f + lane], 0.0f);
    atomicAdd(&sums[(size_t)g * 64 + f + lane], v);
  }
  if (lane == 0) atomicAdd(&cnt[g], 1.0f);
}

// ---------------------------------------------------------------------------
// Classifier: hg = sums/cnt; 4 stacked linears (64->18->12->6->2), no acts.
// One thread per graph (512 total) — ~1.4k MACs each, trivial.
// ---------------------------------------------------------------------------
__global__ void classify_kernel(const float* __restrict__ sums,
                                const float* __restrict__ cnt,
                                const float* __restrict__ Wc1, const float* __restrict__ bc1,
                                const float* __restrict__ Wc2, const float* __restrict__ bc2,
                                const float* __restrict__ Wc3, const float* __restrict__ bc3,
                                const float* __restrict__ Wc4, const float* __restrict__ bc4,
                                float* __restrict__ out) {
  int g = blockIdx.x * blockDim.x + threadIdx.x;
  if (g >= N_GRAPHS) return;
  float inv = 1.0f / fmaxf(cnt[g], 1.0f);

  float h[64];
#pragma unroll
  for (int k = 0; k < 64; ++k) h[k] = sums[(size_t)g * 64 + k] * inv;

  float a1[18];
  for (int j = 0; j < 18; ++j) {
    float s = bc1[j];
    for (int k = 0; k < 64; ++k) s += h[k] * Wc1[k * 18 + j];
    a1[j] = s;
  }
  float a2[12];
  for (int j = 0; j < 12; ++j) {
    float s = bc2[j];
    for (int k = 0; k < 18; ++k) s += a1[k] * Wc2[k * 12 + j];
    a2[j] = s;
  }
  float a3[6];
  for (int j = 0; j < 6; ++j) {
    float s = bc3[j];
    for (int k = 0; k < 12; ++k) s += a2[k] * Wc3[k * 6 + j];
    a3[j] = s;
  }
  for (int j = 0; j < 2; ++j) {
    float s = bc4[j];
    for (int k = 0; k < 6; ++k) s += a3[k] * Wc4[k * 2 + j];
    out[(size_t)g * 2 + j] = s;
  }
}

// ---------------------------------------------------------------------------
extern "C" void kernel_launch(void* const* d_in, const int* in_sizes, int n_in,
                              void* d_out, int out_size, void* d_ws, size_t ws_size,
                              hipStream_t stream) {
  const float* x   = (const float*)d_in[0];
  const int*   src = (const int*)  d_in[1];
  const int*   dst = (const int*)  d_in[2];
  const int*   gid = (const int*)  d_in[3];
  const float* W1  = (const float*)d_in[4];
  const float* b1  = (const float*)d_in[5];
  const float* W2  = (const float*)d_in[6];
  const float* b2  = (const float*)d_in[7];
  const float* Wc1 = (const float*)d_in[8];
  const float* bc1 = (const float*)d_in[9];
  const float* Wc2 = (const float*)d_in[10];
  const float* bc2 = (const float*)d_in[11];
  const float* Wc3 = (const float*)d_in[12];
  const float* bc3 = (const float*)d_in[13];
  const float* Wc4 = (const float*)d_in[14];
  const float* bc4 = (const float*)d_in[15];
  float* out = (float*)d_out;

  // Workspace layout (f32 elements)
  float* ws    = (float*)d_ws;
  float* rsOut = ws;                                   // 100000 (deg -> rsqrt in place)
  float* rsIn  = ws + N_NODES;                         // 100000
  float* agg1  = ws + 2 * (size_t)N_NODES;             // 100000*128
  float* h1    = agg1 + (size_t)N_NODES * 128;         // 100000*256
  float* t2    = h1   + (size_t)N_NODES * 256;         // 100000*64
  float* agg2  = t2   + (size_t)N_NODES * 64;          // 100000*64
  float* sums  = agg2 + (size_t)N_NODES * 64;          // 512*64
  float* cnt   = sums + (size_t)N_GRAPHS * 64;         // 512

  // Zero the accumulation buffers (deterministic per call, capture-safe)
  hipMemsetAsync(rsOut, 0, 2 * (size_t)N_NODES * sizeof(float), stream);
  hipMemsetAsync(agg1,  0, (size_t)N_NODES * 128 * sizeof(float), stream);
  hipMemsetAsync(agg2,  0, (size_t)N_NODES * 64  * sizeof(float), stream);
  hipMemsetAsync(sums,  0, ((size_t)N_GRAPHS * 64 + N_GRAPHS) * sizeof(float), stream);

  const int BLK = 256;

  // 1) degrees
  deg_kernel<<<(N_EDGES + BLK - 1) / BLK, BLK, 0, stream>>>(src, dst, rsOut, rsIn, N_EDGES);
  // 2) deg -> clip^-1/2 (both arrays contiguous)
  rsqrt_kernel<<<(2 * N_NODES + BLK - 1) / BLK, BLK, 0, stream>>>(rsOut, 2 * N_NODES);
  // 3) layer-1 SpMM with fused deg_out^-1/2 scale: agg1 = segsum((x*rsOut)[src] -> dst)
  spmm_scatter128<<<N_EDGES / (BLK / 32), BLK, 0, stream>>>(x, src, dst, rsOut, agg1, N_EDGES);
  // 4) layer-1 GEMM + epilogue: h1 = relu((agg1 @ W1) * rsIn + b1)   [WMMA f32, 16x64/wave]
  {
    int strips = (N_NODES / 16) * (256 / 64);           // 25000 strips
    gemm_wmma_f32_x4<<<(strips + 7) / 8, BLK, 0, stream>>>(agg1, W1, h1, rsIn, b1, 1,
                                                           N_NODES, 128, 256);
  }
  // 5) layer-2 GEMM with fused input row-scale (as output row-scale):
  //    t2 = ((h1*rsOut) @ W2) = (h1 @ W2) * rsOut      [WMMA f32, 16x64/wave]
  {
    int strips = (N_NODES / 16) * (64 / 64);            // 6250 strips
    gemm_wmma_f32_x4<<<(strips + 7) / 8, BLK, 0, stream>>>(h1, W2, t2, rsOut, nullptr, 0,
                                                           N_NODES, 256, 64);
  }
  // 6) layer-2 SpMM: agg2 = segsum(t2[src] -> dst)
  spmm_scatter64<<<N_EDGES / (BLK / 32), BLK, 0, stream>>>(t2, src, dst, agg2, N_EDGES);
  // 7) mean-pool with fused layer-2 epilogue relu(agg2*rsIn + b2)
  pool_kernel<<<(N_NODES * 32 + BLK - 1) / BLK, BLK, 0, stream>>>(agg2, rsIn, b2, gid,
                                                                  sums, cnt, N_NODES);
  // 8) per-graph mean + 4 stacked linear layers -> out (512 x 2)
  classify_kernel<<<(N_GRAPHS + BLK - 1) / BLK, BLK, 0, stream>>>(
      sums, cnt, Wc1, bc1, Wc2, bc2, Wc3, bc3, Wc4, bc4, out);
}